// SEWaveFoward_51883204935989
// MI455X (gfx1250) — compile-verified
//
#include <hip/hip_runtime.h>

typedef __attribute__((ext_vector_type(2))) float v2f;
typedef __attribute__((ext_vector_type(8))) float v8f;

#define NXg 512
#define NNg (NXg * NXg)
#define NB  4
#define NM  256
#define NT  256

// 4th-order Laplacian coefficients (already divided by 12)
__device__ __forceinline__ float stencil_coef(int d) {
    d = d < 0 ? -d : d;
    if (d == 0) return -30.0f / 12.0f;
    if (d == 1) return  16.0f / 12.0f;
    if (d == 2) return  -1.0f / 12.0f;
    return 0.0f;
}

// Zero ping-pong fields, build vel2 = (dt*xp/dx)^2 for the padded grid,
// and clear the measurement map.
__global__ void __launch_bounds__(256)
init_field_kernel(const float* __restrict__ x,
                  float* __restrict__ pA, float* __restrict__ pB,
                  float* __restrict__ vel2, int* __restrict__ map) {
    int idx = blockIdx.x * 256 + threadIdx.x;      // 0 .. NB*NNg-1
    pA[idx] = 0.0f;
    pB[idx] = 0.0f;
    int b    = idx >> 18;                          // / NNg
    int cell = idx & (NNg - 1);
    int u = cell >> 9;
    int v = cell & (NXg - 1);
    float val;
    if (u < 55 || u >= 457 || v < 55 || v >= 457)
        val = 1e-6f;                               // absorbing ring
    else if (u < 128 || u >= 384 || v < 128 || v >= 384)
        val = 1.5f;                                // background ring
    else
        val = x[b * 65536 + (u - 128) * 256 + (v - 128)];
    float w = (2e-5f * val) / 1e-4f;               // dt*xp/dx, f32 order as reference
    vel2[idx] = w * w;
    if (b == 0) map[cell] = -1;
}

// Build measurement map (double-precision trig to match numpy index truncation),
// the 256-sample source wavelet, and the banded-weight WMMA operands in the
// exact per-lane v2f layout (wtab[k*32+lane] = {W[4k+2hi][row], W[4k+2hi+1][row]}).
__global__ void init_meas_kernel(int* __restrict__ map, float* __restrict__ src,
                                 v2f* __restrict__ wtab) {
    int m = threadIdx.x;                           // 256 threads
    double th = 2.0 * 3.14159265358979323846 * (double)m / 256.0;
    int mx = (int)(256.0 + 200.0 * cos(th));
    int my = (int)(256.0 + 200.0 * sin(th));
    map[mx * NXg + my] = m;
    float time = (float)m * 2e-5f;
    float s = sinf((float)(2.0 * 3.14159265358979323846 * 500.0) * time);
    float d = time - 0.002f;
    src[m] = s * expf(-(d * d) / 2e-6f);

    if (m < 5 * 32) {
        int k    = m >> 5;
        int lane = m & 31;
        int row  = lane & 15;
        int hi   = lane >> 4;
        int kb   = 4 * k + 2 * hi;
        v2f w;
        w[0] = stencil_coef(kb + 0 - 2 - row);
        w[1] = stencil_coef(kb + 1 - 2 - row);
        wtab[m] = w;
    }
}

// One time step. One wave32 owns a 16x16 tile; both 1-D stencils are done as
// banded 16x16x20 matmuls via V_WMMA_F32_16X16X4_F32 (5 K-chunks per direction,
// 10 WMMAs total), accumulating nab_x + nab_y into one f32 accumulator.
__global__ void __launch_bounds__(256)
wave_step_kernel(const float* __restrict__ p0, const float* __restrict__ p1,
                 float* __restrict__ pn, const float* __restrict__ vel2,
                 const int* __restrict__ map, const float* __restrict__ src,
                 const v2f* __restrict__ wtab,
                 const float* __restrict__ se, float* __restrict__ out, int t) {
    int lane = threadIdx.x & 31;
    int wave = threadIdx.x >> 5;
    int tile = blockIdx.x * 8 + wave;              // 4096 tiles total
    int b    = tile >> 10;                         // batch
    int tl   = tile & 1023;
    int ti   = (tl >> 5) << 4;                     // tile row origin
    int tj   = (tl & 31) << 4;                     // tile col origin
    int row  = lane & 15;
    int hi   = lane >> 4;

    const float* p1b = p1   + b * NNg;
    const float* p0b = p0   + b * NNg;
    const float* vlb = vel2 + b * NNg;
    float*       pnb = pn   + b * NNg;

    v8f acc = {};                                  // nab_x + nab_y accumulator

#pragma unroll
    for (int k = 0; k < 5; ++k) {
        int kb = 4 * k + 2 * hi;                   // K-pair base within 20-wide band
        // banded weight pair, precomputed in the per-lane WMMA operand layout
        v2f w = wtab[k * 32 + lane];

        // y-direction: A = tile rows with halo (16x4 f32 layout: lane=M,
        // VGPR0/1 = K, lanes 16-31 = K+2 -> one 8B load), B = weights.
        int colbase = (tj - 2 + kb) & (NXg - 1);   // even, never wraps mid-pair
        v2f a = *(const v2f*)(p1b + (ti + row) * NXg + colbase);
        acc = __builtin_amdgcn_wmma_f32_16x16x4_f32(false, a, false, w,
                                                    (short)0, acc, false, false);

        // x-direction: A = weights, B = halo rows (lane holds K=kb+v, N=row).
        int r0 = (ti - 2 + kb) & (NXg - 1);
        int r1 = (r0 + 1) & (NXg - 1);
        v2f bx;
        bx[0] = p1b[r0 * NXg + tj + row];
        bx[1] = p1b[r1 * NXg + tj + row];
        acc = __builtin_amdgcn_wmma_f32_16x16x4_f32(false, w, false, bx,
                                                    (short)0, acc, false, false);
    }

    // SRC_COEF = (bg*dt/dx)^2 = 0.09 (computed in double like the reference)
    float inj = (float)((1.5 * 2e-5 / 1e-4) * (1.5 * 2e-5 / 1e-4)) * src[t];

    // Epilogue on documented C/D layout: VGPR v -> M = v + 8*hi, N = row.
#pragma unroll
    for (int v = 0; v < 8; ++v) {
        int idx = (ti + v + 8 * hi) * NXg + tj + row;
        float res = 2.0f * p1b[idx] - p0b[idx] + vlb[idx] * acc[v];
        int m = map[idx];
        if (m >= 0) {
            if ((m & 3) == 0) res += inj * se[m >> 2];   // source at TRANS = MEAS[::4]
            out[(b << 16) + (m << 8) + t] = res;         // (B, nm, nt) layout
        }
        pnb[idx] = res;
    }
}

extern "C" void kernel_launch(void* const* d_in, const int* in_sizes, int n_in,
                              void* d_out, int out_size, void* d_ws, size_t ws_size,
                              hipStream_t stream) {
    const float* x  = (const float*)d_in[0];   // (4,1,256,256) f32
    const float* se = (const float*)d_in[1];   // (64,) f32
    float* out = (float*)d_out;                // (4,1,256,256) f32

    float* wsf  = (float*)d_ws;
    float* pA   = wsf;                               // 4*512*512 floats
    float* pB   = pA + (size_t)NB * NNg;             // 4*512*512 floats
    float* vel2 = pB + (size_t)NB * NNg;             // 4*512*512 floats
    int*   map  = (int*)(vel2 + (size_t)NB * NNg);   // 512*512 ints
    float* src  = (float*)(map + NNg);               // 256 floats
    v2f*   wtab = (v2f*)(src + NT);                  // 160 v2f = 320 floats

    init_field_kernel<<<(NB * NNg) / 256, 256, 0, stream>>>(x, pA, pB, vel2, map);
    init_meas_kernel<<<1, 256, 0, stream>>>(map, src, wtab);

    float* pprev = pA;   // p0
    float* pcur  = pB;   // p1
    for (int t = 0; t < NT; ++t) {
        // write new field over the old p0 buffer (each cell is read at its own
        // index only by its owning lane before being overwritten)
        wave_step_kernel<<<(NB * 1024) / 8, 256, 0, stream>>>(
            pprev, pcur, pprev, vel2, map, src, wtab, se, out, t);
        float* tmp = pprev; pprev = pcur; pcur = tmp;
    }
}